// Decoder_Cross_Projector_3143916060821
// MI455X (gfx1250) — compile-verified
//
#include <hip/hip_runtime.h>

typedef __attribute__((ext_vector_type(2))) float v2f;
typedef __attribute__((ext_vector_type(4))) float v4f;
typedef __attribute__((ext_vector_type(8))) float v8f;
typedef __attribute__((ext_vector_type(4))) int   v4i;

#define K_DIM 1024      // HIDDEN
#define N_DIM 16384     // out_dim
#define MT 128
#define NT 128
#define KT 16
#define NKT (K_DIM / KT)   // 64 K-steps
#define SA 20           // padded LDS row stride for A (floats)
#define SB 136          // padded LDS row stride for B (floats)

// out layout: stack([k_out, v_out]) -> [2][4][128][2048][64] fp32
#define HEAD_STRIDE   (2048 * 64)
#define BATCH_STRIDE  (128 * 2048 * 64)
#define KV_HALF       ((size_t)4 * 128 * 2048 * 64)

#define AS1 __attribute__((address_space(1)))
#define AS3 __attribute__((address_space(3)))

#if defined(__has_builtin)
#  if __has_builtin(__builtin_amdgcn_global_load_async_to_lds_b128)
#    define USE_ASYNC_LDS 1
#  endif
#endif
#ifndef USE_ASYNC_LDS
#  define USE_ASYNC_LDS 0
#endif

#if USE_ASYNC_LDS
__device__ __forceinline__ void async_cp16(const float* g, float* l) {
    // GLOBAL_LOAD_ASYNC_TO_LDS_B128: memory -> LDS, tracked by ASYNCcnt
    __builtin_amdgcn_global_load_async_to_lds_b128(
        (AS1 v4i*)(AS1 void*)(g),
        (AS3 v4i*)(AS3 void*)(l),
        /*offset=*/0, /*cpol=*/0);
}
__device__ __forceinline__ void wait_async0() {
#  if __has_builtin(__builtin_amdgcn_s_wait_asynccnt)
    __builtin_amdgcn_s_wait_asynccnt(0);
#  else
    asm volatile("s_wait_asynccnt 0" ::: "memory");
#  endif
}
#endif

// One 128x128x16 K-step of WMMA work for this wave (2x4 accumulators).
__device__ __forceinline__ void mma_tile(const float* __restrict__ As,
                                         const float* __restrict__ Bs,
                                         v8f acc[2][4],
                                         int wm, int wn, int l16, int half)
{
    #pragma unroll
    for (int k4 = 0; k4 < KT; k4 += 4) {
        v2f af[2];
        v2f bf[4];
        #pragma unroll
        for (int mi = 0; mi < 2; ++mi) {
            // A 16x4 frag: lane holds M=l16, K = k4 + 2*half + {0,1}
            const float* p = As + (wm * 32 + mi * 16 + l16) * SA + k4 + 2 * half;
            af[mi] = *(const v2f*)p;
        }
        #pragma unroll
        for (int nj = 0; nj < 4; ++nj) {
            // B 4x16 frag: v0 = row (k4+2*half), v1 = next row, col = n
            const float* p = Bs + (k4 + 2 * half) * SB + wn * 64 + nj * 16 + l16;
            v2f bv;
            bv.x = p[0];
            bv.y = p[SB];
            bf[nj] = bv;
        }
        #pragma unroll
        for (int mi = 0; mi < 2; ++mi)
            #pragma unroll
            for (int nj = 0; nj < 4; ++nj)
                acc[mi][nj] = __builtin_amdgcn_wmma_f32_16x16x4_f32(
                    false, af[mi], false, bf[nj],
                    (short)0, acc[mi][nj], false, false);
    }
}

__global__ void __launch_bounds__(256)
dcp_kernel(const float* __restrict__ node,
           const float* __restrict__ mass,
           const float* __restrict__ W,
           const float* __restrict__ bias,
           float* __restrict__ out)
{
    __shared__ float As[2][MT * SA];   // double-buffered [m][k], k contiguous
    __shared__ float Bs[2][KT * SB];   // double-buffered [k][n], n contiguous

    const int tid  = threadIdx.x;
    const int lane = tid & 31;
    const int wave = tid >> 5;
    const int l16  = lane & 15;
    const int half = lane >> 4;
    const int wm   = wave & 3;      // 4 waves along M
    const int wn   = wave >> 2;     // 2 waves along N

    const int tileN = (int)(blockIdx.x & 127) * NT;
    const int tileM = (int)(blockIdx.x >> 7) * MT;

    // staging assignments: 4x b128 per thread per K-step
    const int arow = tid >> 2;            // 0..63 (+64)
    const int aseg = (tid & 3) * 4;       // 0,4,8,12
    const int brow = tid >> 5;            // 0..7 (+8)
    const int bseg = (tid & 31) * 4;      // 0..124

    const float* gA0 = node + (size_t)(tileM + arow)      * K_DIM + aseg;
    const float* gA1 = node + (size_t)(tileM + arow + 64) * K_DIM + aseg;
    const float* gB0 = W    + (size_t)brow       * N_DIM + tileN + bseg;
    const float* gB1 = W    + (size_t)(brow + 8) * N_DIM + tileN + bseg;

    const int lA0 = arow * SA + aseg;
    const int lA1 = (arow + 64) * SA + aseg;
    const int lB0 = brow * SB + bseg;
    const int lB1 = (brow + 8) * SB + bseg;

    v8f acc[2][4] = {};

#if USE_ASYNC_LDS
    // ---- async double-buffered pipeline: one barrier per K-step ----
    async_cp16(gA0, &As[0][lA0]);
    async_cp16(gA1, &As[0][lA1]);
    async_cp16(gB0, &Bs[0][lB0]);
    async_cp16(gB1, &Bs[0][lB1]);

    for (int i = 0; i < NKT; ++i) {
        const int buf = i & 1;
        wait_async0();          // my copies into As/Bs[buf] have landed
        __syncthreads();        // everyone's copies landed; prev reads of buf^1 done
        if (i + 1 < NKT) {
            const int nb = buf ^ 1;
            const int kt = (i + 1) * KT;
            async_cp16(gA0 + kt,                  &As[nb][lA0]);
            async_cp16(gA1 + kt,                  &As[nb][lA1]);
            async_cp16(gB0 + (size_t)kt * N_DIM,  &Bs[nb][lB0]);
            async_cp16(gB1 + (size_t)kt * N_DIM,  &Bs[nb][lB1]);
        }
        mma_tile(&As[buf][0], &Bs[buf][0], acc, wm, wn, l16, half);
    }
#else
    // ---- fallback: register-staged double buffer, one barrier per K-step ----
    v4f ra0 = *(const v4f*)gA0;
    v4f ra1 = *(const v4f*)gA1;
    v4f rb0 = *(const v4f*)gB0;
    v4f rb1 = *(const v4f*)gB1;

    for (int i = 0; i < NKT; ++i) {
        const int buf = i & 1;
        *(v4f*)&As[buf][lA0] = ra0;
        *(v4f*)&As[buf][lA1] = ra1;
        *(v4f*)&Bs[buf][lB0] = rb0;
        *(v4f*)&Bs[buf][lB1] = rb1;
        if (i + 1 < NKT) {
            const int kt = (i + 1) * KT;
            ra0 = *(const v4f*)(gA0 + kt);
            ra1 = *(const v4f*)(gA1 + kt);
            rb0 = *(const v4f*)(gB0 + (size_t)kt * N_DIM);
            rb1 = *(const v4f*)(gB1 + (size_t)kt * N_DIM);
        }
        __syncthreads();        // flushes my ds stores; all waves' buf ready
        mma_tile(&As[buf][0], &Bs[buf][0], acc, wm, wn, l16, half);
    }
#endif

    // ---------------- epilogue: bias + RoPE(K) + transpose scatter ----------------
    const int o0 = tileN + wn * 64;       // head-aligned column base (64-wide)
    const int c  = o0 >> 6;               // <128 => K cache, >=128 => V cache

    float bv[4];
    #pragma unroll
    for (int nj = 0; nj < 4; ++nj)
        bv[nj] = bias[o0 + nj * 16 + l16];

    if (c < 128) {
        const size_t headbase = (size_t)c * HEAD_STRIDE;
        float invf[2];
        #pragma unroll
        for (int q = 0; q < 2; ++q) {
            const int j = q * 16 + l16;   // 0..31
            invf[q] = expf(-9.2103403719761836f * (float)j / 32.0f);
        }
        #pragma unroll
        for (int mi = 0; mi < 2; ++mi) {
            #pragma unroll
            for (int r = 0; r < 8; ++r) {
                const int m    = tileM + wm * 32 + mi * 16 + r + 8 * half;
                const int bidx = m >> 11;
                const int nrow = m & 2047;
                const float ms = mass[m];
                const size_t rowbase =
                    (size_t)bidx * BATCH_STRIDE + headbase + (size_t)nrow * 64;
                #pragma unroll
                for (int q = 0; q < 2; ++q) {
                    const float ang = ms * invf[q];
                    const float s  = sinf(ang);
                    const float co = cosf(ang);
                    const float x0 = acc[mi][q][r]     + bv[q];
                    const float x1 = acc[mi][q + 2][r] + bv[q + 2];
                    const int d = q * 16 + l16;
                    __builtin_nontemporal_store(x0 * co - x1 * s,  &out[rowbase + d]);
                    __builtin_nontemporal_store(x0 * s  + x1 * co, &out[rowbase + d + 32]);
                }
            }
        }
    } else {
        const size_t headbase = KV_HALF + (size_t)(c - 128) * HEAD_STRIDE;
        #pragma unroll
        for (int mi = 0; mi < 2; ++mi) {
            #pragma unroll
            for (int r = 0; r < 8; ++r) {
                const int m    = tileM + wm * 32 + mi * 16 + r + 8 * half;
                const int bidx = m >> 11;
                const int nrow = m & 2047;
                const size_t rowbase =
                    (size_t)bidx * BATCH_STRIDE + headbase + (size_t)nrow * 64;
                #pragma unroll
                for (int nj = 0; nj < 4; ++nj)
                    __builtin_nontemporal_store(acc[mi][nj][r] + bv[nj],
                                                &out[rowbase + nj * 16 + l16]);
            }
        }
    }
}

extern "C" void kernel_launch(void* const* d_in, const int* in_sizes, int n_in,
                              void* d_out, int out_size, void* d_ws, size_t ws_size,
                              hipStream_t stream) {
    const float* node = (const float*)d_in[0];   // [4,2048,1024]
    const float* mass = (const float*)d_in[1];   // [4,2048]
    const float* W    = (const float*)d_in[2];   // [1024,16384]
    const float* bias = (const float*)d_in[3];   // [16384]
    float* out = (float*)d_out;                  // [2,4,128,2048,64]

    dim3 grid(64 * 128);   // (M/128) * (N/128), N-major for A-panel L2 reuse
    dim3 block(256);
    hipLaunchKernelGGL(dcp_kernel, grid, block, 0, stream,
                       node, mass, W, bias, out);
}